// VariationalGCNEncoder_12472585028061
// MI455X (gfx1250) — compile-verified
//
#include <hip/hip_runtime.h>
#include <hip/hip_bf16.h>

typedef __attribute__((ext_vector_type(2))) float v2f;
typedef __attribute__((ext_vector_type(8))) float v8f;

// ---------------------------------------------------------------------------
// Elementwise helpers
// ---------------------------------------------------------------------------
__global__ void fill_f32(float* __restrict__ p, float v, long n) {
    long i = (long)blockIdx.x * blockDim.x + threadIdx.x;
    if (i < n) p[i] = v;
}

// out[i*C + c] = b[c]  (C is a power of two)
__global__ void init_bias(float* __restrict__ out, const float* __restrict__ b,
                          long total, int Cmask) {
    long i = (long)blockIdx.x * blockDim.x + threadIdx.x;
    if (i < total) out[i] = b[i & Cmask];
}

// deg[col[e]] += 1  (deg pre-filled with 1.0 for the self-loop)
__global__ void degree_kernel(const int* __restrict__ col, float* __restrict__ deg, int E) {
    int i = blockIdx.x * blockDim.x + threadIdx.x;
    if (i < E) atomicAdd(deg + col[i], 1.0f);
}

// in-place deg -> deg^{-1/2} (deg >= 1 always, no zero guard needed)
__global__ void rsqrt_kernel(float* __restrict__ d, int n) {
    int i = blockIdx.x * blockDim.x + threadIdx.x;
    if (i < n) d[i] = rsqrtf(d[i]);
}

// h = relu(h + b[c]), C = 128
__global__ void relu_bias_kernel(float* __restrict__ h, const float* __restrict__ b, long n) {
    long i = (long)blockIdx.x * blockDim.x + threadIdx.x;
    if (i < n) {
        float v = h[i] + b[i & 127];
        h[i] = v > 0.0f ? v : 0.0f;
    }
}

// ---------------------------------------------------------------------------
// FP32 WMMA GEMM:  Y[nRows x nCols] = A[nRows x 128] * W[128 x nCols]
// One wave per 16x16 output tile; 8 waves/block cover 128 rows x 16 cols.
// Weight slice W[:, colTile:colTile+16] staged in LDS (8 KB).
// A-fragment layout (V_WMMA_F32_16X16X4_F32): lane m = lane&15 is row M;
// lanes 0-15 hold K = k0, k0+1; lanes 16-31 hold K = k0+2, k0+3.
// B-fragment mirrors the K split; C/D: VGPR g -> row g (+8 for lanes>=16).
// nRows must be a multiple of 16 (50000 = 3125*16), nCols multiple of 16.
// ---------------------------------------------------------------------------
__global__ void gemm_wmma_k128(const float* __restrict__ A, const float* __restrict__ W,
                               float* __restrict__ Y, int nRows, int nCols) {
    __shared__ float Wl[128 * 16];

    const int colTile = blockIdx.y * 16;
    // cooperative stage of the 128x16 weight slice
    for (int i = threadIdx.x; i < 128 * 16; i += blockDim.x) {
        int k = i >> 4, c = i & 15;
        Wl[i] = W[k * nCols + colTile + c];
    }
    __syncthreads();

    const int wave = threadIdx.x >> 5;
    const int lane = threadIdx.x & 31;
    const int row0 = blockIdx.x * 128 + wave * 16;
    if (row0 + 16 > nRows) return;      // wave-uniform: EXEC stays all-ones

    const int m    = lane & 15;         // row within tile / col within tile
    const int kb   = (lane >> 4) << 1;  // 0 for lanes 0-15, 2 for lanes 16-31
    const float* Arow = A + (long)(row0 + m) * 128;

    v8f acc = {};
#pragma unroll
    for (int k0 = 0; k0 < 128; k0 += 4) {
        v2f a, b;
        a.x = Arow[k0 + kb];
        a.y = Arow[k0 + kb + 1];
        b.x = Wl[(k0 + kb) * 16 + m];
        b.y = Wl[(k0 + kb + 1) * 16 + m];
        acc = __builtin_amdgcn_wmma_f32_16x16x4_f32(
            /*neg_a=*/false, a, /*neg_b=*/false, b,
            /*c_mod=*/(short)0, acc, /*reuse_a=*/false, /*reuse_b=*/false);
    }

    const int radd = (lane >> 4) << 3;  // +8 rows for lanes 16-31
#pragma unroll
    for (int g = 0; g < 8; ++g)
        Y[(long)(row0 + radd + g) * nCols + colTile + m] = acc[g];
}

// ---------------------------------------------------------------------------
// Edge scatter, 128 channels: one wave per edge, 4 channels (float4) per lane.
// Virtual edges e in [E, E+N) are the self-loops (row = col = e - E).
// ---------------------------------------------------------------------------
__global__ void aggregate128(const float* __restrict__ src,
                             const int* __restrict__ rowI, const int* __restrict__ colI,
                             const float* __restrict__ dis,
                             float* __restrict__ dst, int E, int N) {
    long w = ((long)blockIdx.x * blockDim.x + threadIdx.x) >> 5;
    int lane = threadIdx.x & 31;
    if (w >= (long)E + N) return;
    int r, c;
    if (w < E) { r = rowI[w]; c = colI[w]; } else { r = (int)(w - E); c = r; }
    float nrm = dis[r] * dis[c];
    const float4 v = *reinterpret_cast<const float4*>(src + (long)r * 128 + lane * 4);
    float* d = dst + (long)c * 128 + lane * 4;
    atomicAdd(d + 0, v.x * nrm);
    atomicAdd(d + 1, v.y * nrm);
    atomicAdd(d + 2, v.z * nrm);
    atomicAdd(d + 3, v.w * nrm);
}

// ---------------------------------------------------------------------------
// Edge scatter for mu and logstd together, 64 channels each:
// one wave per edge, 2 channels (float2) per lane per head.
// ---------------------------------------------------------------------------
__global__ void aggregate64x2(const float* __restrict__ tmu, const float* __restrict__ tls,
                              const int* __restrict__ rowI, const int* __restrict__ colI,
                              const float* __restrict__ dis,
                              float* __restrict__ omu, float* __restrict__ ols,
                              int E, int N) {
    long w = ((long)blockIdx.x * blockDim.x + threadIdx.x) >> 5;
    int lane = threadIdx.x & 31;
    if (w >= (long)E + N) return;
    int r, c;
    if (w < E) { r = rowI[w]; c = colI[w]; } else { r = (int)(w - E); c = r; }
    float nrm = dis[r] * dis[c];
    int ch = lane * 2;
    const float2 vm = *reinterpret_cast<const float2*>(tmu + (long)r * 64 + ch);
    const float2 vl = *reinterpret_cast<const float2*>(tls + (long)r * 64 + ch);
    float* dm = omu + (long)c * 64 + ch;
    float* dl = ols + (long)c * 64 + ch;
    atomicAdd(dm + 0, vm.x * nrm);
    atomicAdd(dm + 1, vm.y * nrm);
    atomicAdd(dl + 0, vl.x * nrm);
    atomicAdd(dl + 1, vl.y * nrm);
}

// ---------------------------------------------------------------------------
extern "C" void kernel_launch(void* const* d_in, const int* in_sizes, int n_in,
                              void* d_out, int out_size, void* d_ws, size_t ws_size,
                              hipStream_t stream) {
    (void)n_in; (void)out_size; (void)ws_size;
    const float* x   = (const float*)d_in[0];
    const int*   ei  = (const int*)d_in[1];
    const float* W1  = (const float*)d_in[2];
    const float* b1  = (const float*)d_in[3];
    const float* Wmu = (const float*)d_in[4];
    const float* bmu = (const float*)d_in[5];
    const float* Wls = (const float*)d_in[6];
    const float* bls = (const float*)d_in[7];

    const int N = in_sizes[0] / 128;
    const int E = in_sizes[1] / 2;
    const int* rowI = ei;       // edge_index[0] = source
    const int* colI = ei + E;   // edge_index[1] = target

    // Workspace layout (floats): dis | h_pre (reused as t_mu/t_ls) | h_agg
    float* ws     = (float*)d_ws;
    float* degdis = ws;                          // N
    float* h_pre  = ws + N;                      // N*128
    float* h_agg  = h_pre + (size_t)N * 128;     // N*128
    float* t_mu   = h_pre;                       // N*64  (reuse after aggregate128)
    float* t_ls   = h_pre + (size_t)N * 64;      // N*64
    float* out_mu = (float*)d_out;               // N*64
    float* out_ls = out_mu + (size_t)N * 64;     // N*64

    const int BT = 256;
    const long nh = (long)N * 128;
    const long no = (long)N * 64;
    const long edgeThreads = ((long)E + N) * 32;

    // 1) norm precompute (shared by all three convs, like cached=True)
    fill_f32<<<dim3((unsigned)((N + BT - 1) / BT)), BT, 0, stream>>>(degdis, 1.0f, N);
    fill_f32<<<dim3((unsigned)((nh + BT - 1) / BT)), BT, 0, stream>>>(h_agg, 0.0f, nh);
    degree_kernel<<<dim3((unsigned)((E + BT - 1) / BT)), BT, 0, stream>>>(colI, degdis, E);
    rsqrt_kernel<<<dim3((unsigned)((N + BT - 1) / BT)), BT, 0, stream>>>(degdis, N);

    // 2) layer 1: h_pre = x @ W1  (WMMA fp32)
    gemm_wmma_k128<<<dim3((unsigned)((N + 127) / 128), 8), BT, 0, stream>>>(x, W1, h_pre, N, 128);
    // scatter: h_agg[col] += h_pre[row] * norm
    aggregate128<<<dim3((unsigned)((edgeThreads + BT - 1) / BT)), BT, 0, stream>>>(
        h_pre, rowI, colI, degdis, h_agg, E, N);
    // h = relu(h_agg + b1)
    relu_bias_kernel<<<dim3((unsigned)((nh + BT - 1) / BT)), BT, 0, stream>>>(h_agg, b1, nh);

    // 3) heads: t_mu = h @ Wmu, t_ls = h @ Wls  (WMMA fp32)
    gemm_wmma_k128<<<dim3((unsigned)((N + 127) / 128), 4), BT, 0, stream>>>(h_agg, Wmu, t_mu, N, 64);
    gemm_wmma_k128<<<dim3((unsigned)((N + 127) / 128), 4), BT, 0, stream>>>(h_agg, Wls, t_ls, N, 64);

    // 4) out = bias, then scatter-add both heads into d_out
    init_bias<<<dim3((unsigned)((no + BT - 1) / BT)), BT, 0, stream>>>(out_mu, bmu, no, 63);
    init_bias<<<dim3((unsigned)((no + BT - 1) / BT)), BT, 0, stream>>>(out_ls, bls, no, 63);
    aggregate64x2<<<dim3((unsigned)((edgeThreads + BT - 1) / BT)), BT, 0, stream>>>(
        t_mu, t_ls, rowI, colI, degdis, out_mu, out_ls, E, N);
}